// UnifiedCascadePredictionModel_3642132267419
// MI455X (gfx1250) — compile-verified
//
#include <hip/hip_runtime.h>
#include <hip/hip_bf16.h>

// Problem constants (match reference)
#define BB   4
#define NN   10000
#define EE   160000
#define HH   4
#define CC   32
#define FF   128           // H*C
#define ED   16
#define ETOT (EE + NN)     // 170000
#define MM   (BB * NN)     // 40000

typedef __attribute__((ext_vector_type(2))) float v2f;
typedef __attribute__((ext_vector_type(8))) float v8f;

// ---- monotone float<->uint encoding for atomicMax-based segment max ----
__device__ __forceinline__ unsigned fenc(float x) {
    unsigned b = __float_as_uint(x);
    return (b & 0x80000000u) ? ~b : (b | 0x80000000u);
}
__device__ __forceinline__ float fdec(unsigned k) {
    unsigned b = (k & 0x80000000u) ? (k & 0x7FFFFFFFu) : ~k;
    return __uint_as_float(b);
}
#define NEG_INF_ENC 0x007FFFFFu   // fenc(-inf)

// ---------------------------------------------------------------------------
// Kernel 0: transpose weights so WMMA B-fragments load coalesced.
// Wt[c][f] = W[f][c]   (128x128),  Wet[d][f] = W_edge[f][d]  (16x128)
// ---------------------------------------------------------------------------
__global__ __launch_bounds__(256) void k_transpose(const float* __restrict__ W,
                                                   const float* __restrict__ We,
                                                   float* __restrict__ Wt,
                                                   float* __restrict__ Wet) {
    int tid = blockIdx.x * 256 + threadIdx.x;
    if (tid < FF * FF) {
        int f = tid & (FF - 1), c = tid >> 7;
        Wt[tid] = W[f * FF + c];
    }
    int u = tid - FF * FF;
    if (u >= 0 && u < ED * FF) {
        int f = u & (FF - 1), d = u >> 7;
        Wet[u] = We[f * ED + d];
    }
}

// ---------------------------------------------------------------------------
// Kernel 1: xt = x @ W^T via V_WMMA_F32_16X16X4_F32, fused with
// a_src[m][h] = xt[m][h*32:...]·att_src, a_dst likewise.
// One block = 16 rows of xt, 8 waves each owning a 16-wide column tile.
// ---------------------------------------------------------------------------
__global__ __launch_bounds__(256) void k_node_gemm(const float* __restrict__ x,
                                                   const float* __restrict__ Wt,
                                                   const float* __restrict__ att_src,
                                                   const float* __restrict__ att_dst,
                                                   float* __restrict__ xt,
                                                   float* __restrict__ a_src,
                                                   float* __restrict__ a_dst) {
    __shared__ float lds[16][FF];
    const int wave = threadIdx.x >> 5;
    const int lane = threadIdx.x & 31;
    const int m0 = blockIdx.x * 16;        // 2500 blocks * 16 = 40000 rows
    const int f0 = wave * 16;
    const int lrow  = lane & 15;
    const int khalf = (lane >> 4) << 1;    // 0 or 2

    v8f acc = {};
    const float* xrow = x + (size_t)(m0 + lrow) * FF;
#pragma unroll
    for (int k0 = 0; k0 < FF; k0 += 4) {
        v2f a, b;
        a.x = xrow[k0 + khalf];
        a.y = xrow[k0 + khalf + 1];
        b.x = Wt[(k0 + khalf)     * FF + f0 + lrow];
        b.y = Wt[(k0 + khalf + 1) * FF + f0 + lrow];
        acc = __builtin_amdgcn_wmma_f32_16x16x4_f32(false, a, false, b,
                                                    (short)0, acc, false, false);
    }
    // D layout: lanes 0-15 hold M=v, lanes 16-31 hold M=v+8, N = lane&15
    const int col   = f0 + lrow;
    const int rbase = (lane >> 4) * 8;
#pragma unroll
    for (int v = 0; v < 8; ++v) {
        float val = acc[v];
        xt[(size_t)(m0 + rbase + v) * FF + col] = val;
        lds[rbase + v][col] = val;
    }
    __syncthreads();
    // 16 rows * 4 heads * 2 (src/dst) = 128 length-32 dot products
    if (threadIdx.x < 128) {
        int r = threadIdx.x >> 3, q = threadIdx.x & 7;
        int h = q >> 1, isDst = q & 1;
        const float* att = isDst ? att_dst : att_src;
        float s = 0.f;
#pragma unroll
        for (int c = 0; c < CC; ++c) s += lds[r][h * CC + c] * att[h * CC + c];
        (isDst ? a_dst : a_src)[(size_t)(m0 + r) * HH + h] = s;
    }
}

// ---------------------------------------------------------------------------
// Kernel 2: a_edge[e][h] = (edge_attr @ W_edge^T)[e, h*32:...]·att_edge
// via WMMA (K=16 -> 4 WMMA steps), ea tile reduced in LDS (never stored).
// ---------------------------------------------------------------------------
__global__ __launch_bounds__(256) void k_edge_gemm(const float* __restrict__ eattr,
                                                   const float* __restrict__ Wet,
                                                   const float* __restrict__ att_edge,
                                                   float* __restrict__ a_edge) {
    __shared__ float lds[16][FF];
    const int wave = threadIdx.x >> 5;
    const int lane = threadIdx.x & 31;
    const int e0 = blockIdx.x * 16;        // 10000 blocks * 16 = 160000 edges
    const int f0 = wave * 16;
    const int lrow  = lane & 15;
    const int khalf = (lane >> 4) << 1;

    v8f acc = {};
    const float* arow = eattr + (size_t)(e0 + lrow) * ED;
#pragma unroll
    for (int k0 = 0; k0 < ED; k0 += 4) {
        v2f a, b;
        a.x = arow[k0 + khalf];
        a.y = arow[k0 + khalf + 1];
        b.x = Wet[(k0 + khalf)     * FF + f0 + lrow];
        b.y = Wet[(k0 + khalf + 1) * FF + f0 + lrow];
        acc = __builtin_amdgcn_wmma_f32_16x16x4_f32(false, a, false, b,
                                                    (short)0, acc, false, false);
    }
    const int col   = f0 + lrow;
    const int rbase = (lane >> 4) * 8;
#pragma unroll
    for (int v = 0; v < 8; ++v) lds[rbase + v][col] = acc[v];
    __syncthreads();
    if (threadIdx.x < 64) {   // 16 edges * 4 heads
        int el = threadIdx.x >> 2, h = threadIdx.x & 3;
        float s = 0.f;
#pragma unroll
        for (int c = 0; c < CC; ++c) s += lds[el][h * CC + c] * att_edge[h * CC + c];
        a_edge[(size_t)(e0 + el) * HH + h] = s;
    }
}

// ---------------------------------------------------------------------------
// Kernel 3: init amax (encoded -inf), denom=0, out=0
// ---------------------------------------------------------------------------
__global__ __launch_bounds__(256) void k_init(unsigned* __restrict__ amax,
                                              float* __restrict__ denom,
                                              float4* __restrict__ out4) {
    int tid = blockIdx.x * 256 + threadIdx.x;
    if (tid < NN * BB * HH) { amax[tid] = NEG_INF_ENC; denom[tid] = 0.f; }
    if (tid < (MM * FF) / 4) out4[tid] = make_float4(0.f, 0.f, 0.f, 0.f);
}

// ---------------------------------------------------------------------------
// Kernel 4: alpha = lrelu(a_src[src]+a_dst[dst]+a_edge); segment max via
// atomicMax on monotone-encoded floats. Layout alpha[e][b][h].
// ---------------------------------------------------------------------------
__global__ __launch_bounds__(256) void k_alpha_max(const int* __restrict__ ei,
                                                   const float* __restrict__ a_src,
                                                   const float* __restrict__ a_dst,
                                                   const float* __restrict__ a_edge,
                                                   float* __restrict__ alpha,
                                                   unsigned* __restrict__ amax) {
    int tid = blockIdx.x * 256 + threadIdx.x;       // ETOT*B*H = 2,720,000
    if (tid >= ETOT * BB * HH) return;
    int h = tid & 3;
    int b = (tid >> 2) & 3;
    int e = tid >> 4;
    int s, d; float ae;
    if (e < EE) { s = ei[e]; d = ei[EE + e]; ae = a_edge[(size_t)e * HH + h]; }
    else        { s = d = e - EE; ae = 0.f; }   // self loop, zero edge_attr
    float al = a_src[((size_t)b * NN + s) * HH + h]
             + a_dst[((size_t)b * NN + d) * HH + h] + ae;
    al = (al > 0.f) ? al : 0.2f * al;           // leaky_relu(0.2)
    alpha[tid] = al;
    atomicMax(&amax[((size_t)d * BB + b) * HH + h], fenc(al));
}

// ---------------------------------------------------------------------------
// Kernel 5: a_exp = exp(alpha - amax[dst]); denom segment-sum via atomicAdd
// ---------------------------------------------------------------------------
__global__ __launch_bounds__(256) void k_exp_denom(const int* __restrict__ ei,
                                                   const unsigned* __restrict__ amax,
                                                   float* __restrict__ alpha,
                                                   float* __restrict__ denom) {
    int tid = blockIdx.x * 256 + threadIdx.x;
    if (tid >= ETOT * BB * HH) return;
    int h = tid & 3;
    int b = (tid >> 2) & 3;
    int e = tid >> 4;
    int d = (e < EE) ? ei[EE + e] : (e - EE);
    float m = fdec(amax[((size_t)d * BB + b) * HH + h]);
    float ex = expf(alpha[tid] - m);
    alpha[tid] = ex;
    atomicAdd(&denom[((size_t)d * BB + b) * HH + h], ex);
}

// ---------------------------------------------------------------------------
// Kernel 6: out[b,dst,:] += xt[b,src,:] * (a_exp/(denom+1e-16)).
// One thread per (edge, batch, 4-float group): float4 gather + 4 atomic adds.
// ---------------------------------------------------------------------------
__global__ __launch_bounds__(256) void k_scatter(const int* __restrict__ ei,
                                                 const float* __restrict__ xt,
                                                 const float* __restrict__ alpha,
                                                 const float* __restrict__ denom,
                                                 float* __restrict__ out) {
    long tid = (long)blockIdx.x * 256 + threadIdx.x;  // ETOT*B*32 = 21,760,000
    if (tid >= (long)ETOT * BB * 32) return;
    int j = (int)(tid & 31);           // float4 group: f = 4j..4j+3
    int b = (int)((tid >> 5) & 3);
    int e = (int)(tid >> 7);
    int s, d;
    if (e < EE) { s = ei[e]; d = ei[EE + e]; }
    else        { s = d = e - EE; }
    int h = j >> 3;                    // (4j)/32
    float w  = alpha[((size_t)e * BB + b) * HH + h];
    float dn = denom[((size_t)d * BB + b) * HH + h] + 1e-16f;
    w /= dn;
    const float4 xv = ((const float4*)(xt + ((size_t)b * NN + s) * FF))[j];
    float* op = out + ((size_t)b * NN + d) * FF + j * 4;
    atomicAdd(op + 0, xv.x * w);
    atomicAdd(op + 1, xv.y * w);
    atomicAdd(op + 2, xv.z * w);
    atomicAdd(op + 3, xv.w * w);
}

// ---------------------------------------------------------------------------
// Kernel 7: out += bias (broadcast over [B,N])
// ---------------------------------------------------------------------------
__global__ __launch_bounds__(256) void k_bias(float4* __restrict__ out4,
                                              const float* __restrict__ bias) {
    int tid = blockIdx.x * 256 + threadIdx.x;       // MM*FF/4 = 1,280,000
    if (tid >= (MM * FF) / 4) return;
    float4 bv = ((const float4*)bias)[tid & 31];
    float4 o = out4[tid];
    o.x += bv.x; o.y += bv.y; o.z += bv.z; o.w += bv.w;
    out4[tid] = o;
}

// ---------------------------------------------------------------------------
extern "C" void kernel_launch(void* const* d_in, const int* in_sizes, int n_in,
                              void* d_out, int out_size, void* d_ws, size_t ws_size,
                              hipStream_t stream) {
    const float* x        = (const float*)d_in[0];   // [B,N,128]
    const float* eattr    = (const float*)d_in[1];   // [E,16]
    const float* W        = (const float*)d_in[2];   // [128,128]
    const float* att_src  = (const float*)d_in[3];   // [1,4,32]
    const float* att_dst  = (const float*)d_in[4];
    const float* We       = (const float*)d_in[5];   // [128,16]
    const float* att_edge = (const float*)d_in[6];
    const float* bias     = (const float*)d_in[7];   // [128]
    const int*   ei       = (const int*)d_in[8];     // [2,E]
    float* out = (float*)d_out;                      // [B,N,128]

    // workspace layout (all 256B aligned)
    char* ws = (char*)d_ws;
    size_t off = 0;
    float*    Wt     = (float*)(ws + off); off += (size_t)FF * FF * 4;      //  64 KB
    float*    Wet    = (float*)(ws + off); off += (size_t)ED * FF * 4;      //   8 KB
    float*    xt     = (float*)(ws + off); off += (size_t)MM * FF * 4;      // 20.5 MB
    float*    a_src  = (float*)(ws + off); off += (size_t)MM * HH * 4;      // 640 KB
    float*    a_dst  = (float*)(ws + off); off += (size_t)MM * HH * 4;      // 640 KB
    float*    a_edge = (float*)(ws + off); off += (size_t)EE * HH * 4;      // 2.56 MB
    float*    alpha  = (float*)(ws + off); off += (size_t)ETOT * BB * HH * 4; // 10.9 MB
    unsigned* amax   = (unsigned*)(ws + off); off += (size_t)NN * BB * HH * 4; // 640 KB
    float*    denom  = (float*)(ws + off); off += (size_t)NN * BB * HH * 4;    // 640 KB

    k_transpose<<<(FF * FF + ED * FF + 255) / 256, 256, 0, stream>>>(W, We, Wt, Wet);
    k_node_gemm<<<MM / 16, 256, 0, stream>>>(x, Wt, att_src, att_dst, xt, a_src, a_dst);
    k_edge_gemm<<<EE / 16, 256, 0, stream>>>(eattr, Wet, att_edge, a_edge);
    k_init<<<(MM * FF / 4 + 255) / 256, 256, 0, stream>>>(amax, denom, (float4*)out);
    k_alpha_max<<<(ETOT * BB * HH) / 256, 256, 0, stream>>>(ei, a_src, a_dst, a_edge, alpha, amax);
    k_exp_denom<<<(ETOT * BB * HH) / 256, 256, 0, stream>>>(ei, amax, alpha, denom);
    k_scatter<<<(int)(((long)ETOT * BB * 32) / 256), 256, 0, stream>>>(ei, xt, alpha, denom, out);
    k_bias<<<(MM * FF / 4) / 256, 256, 0, stream>>>((float4*)out, bias);
}